// GCNModel_33397665694652
// MI455X (gfx1250) — compile-verified
//
#include <hip/hip_runtime.h>
#include <hip/hip_bf16.h>

// ---------------------------------------------------------------------------
// GCN (3x GCNConv + BN + ReLU, FC + sigmoid) for MI455X / gfx1250.
// Memory-bound workload (~6 GB traffic -> ~260us floor @ 23.3 TB/s), so the
// GEMMs use exact-precision V_WMMA_F32_16X16X4_F32 (matrix units are idle
// anyway) and the segment-sum uses native f32 global atomics.
// ---------------------------------------------------------------------------

#define GCN_H 256            // hidden/feature width (Fin == H == 256)
#define LDSS  258            // padded LDS row stride (words) to avoid bank conflicts
#define BN_EPS 1e-5f

typedef float v2f __attribute__((ext_vector_type(2)));
typedef float v8f __attribute__((ext_vector_type(8)));

// ---------------------------------------------------------------------------
// Degree kernels
// ---------------------------------------------------------------------------
__global__ void gcn_deg_init(float* __restrict__ deg, int n) {
  int i = blockIdx.x * blockDim.x + threadIdx.x;
  if (i < n) deg[i] = 1.0f;  // self loop
}

__global__ void gcn_deg_accum(const int* __restrict__ dst, float* __restrict__ deg, int e) {
  int i = blockIdx.x * blockDim.x + threadIdx.x;
  if (i < e) unsafeAtomicAdd(&deg[dst[i]], 1.0f);
}

__global__ void gcn_deg_finalize(float* __restrict__ deg, int n) {
  int i = blockIdx.x * blockDim.x + threadIdx.x;
  if (i < n) deg[i] = rsqrtf(deg[i]);   // deg >= 1 always (self loops)
}

// ---------------------------------------------------------------------------
// Zero-fill (float4)
// ---------------------------------------------------------------------------
__global__ void gcn_fill_zero4(float4* __restrict__ p, int n4) {
  int i = blockIdx.x * blockDim.x + threadIdx.x;
  if (i < n4) p[i] = make_float4(0.f, 0.f, 0.f, 0.f);
}

// ---------------------------------------------------------------------------
// GEMM: C[M,256] = A[M,256] * W[256,256], fp32 WMMA 16x16x4.
// Block = 512 threads = 16 waves; block owns 16 rows, wave w owns cols
// [16w, 16w+16). A tile staged in LDS (padded stride).
// ---------------------------------------------------------------------------
__global__ __launch_bounds__(512) void gcn_gemm_wmma(
    const float* __restrict__ A, const float* __restrict__ W,
    float* __restrict__ C) {
  __shared__ float sA[16 * LDSS];
  const int row0 = blockIdx.x * 16;
  const int tid  = threadIdx.x;

  // Stage 16x256 A tile: 1024 float4 loads, 2 per thread, fully coalesced.
  const float4* Ap = (const float4*)(A + (size_t)row0 * GCN_H);
  #pragma unroll
  for (int g = tid; g < 1024; g += 512) {
    float4 v = Ap[g];
    const int r  = g >> 6;
    const int c  = (g & 63) << 2;
    *(float4*)&sA[r * LDSS + c] = v;
  }
  __syncthreads();

  const int wave = tid >> 5;
  const int lane = tid & 31;
  const int half = lane >> 4;            // 0: lanes 0-15, 1: lanes 16-31
  const int l15  = lane & 15;
  const int col  = (wave << 4) + l15;    // output column
  const int koff = half << 1;            // ISA: upper half-wave holds K+2,K+3

  v8f acc = {0.f, 0.f, 0.f, 0.f, 0.f, 0.f, 0.f, 0.f};

  for (int k = 0; k < GCN_H; k += 4) {
    const int kb = k + koff;
    // A fragment (16x4 f32 layout): lane l15 = row M, {K=kb, K=kb+1}
    v2f a = *(const v2f*)&sA[l15 * LDSS + kb];
    // B fragment (4x16): lane l15 = col N, {K=kb, K=kb+1}
    v2f b;
    b.x = W[(size_t)kb * GCN_H + col];
    b.y = W[(size_t)(kb + 1) * GCN_H + col];
    acc = __builtin_amdgcn_wmma_f32_16x16x4_f32(false, a, false, b,
                                                (short)0, acc, false, false);
  }

  // D layout: VGPR v -> row (half? 8+v : v), col = lane&15 within tile.
  const int rbase = row0 + (half << 3);
  #pragma unroll
  for (int i = 0; i < 8; ++i)
    C[(size_t)(rbase + i) * GCN_H + col] = acc[i];
}

// ---------------------------------------------------------------------------
// Edge scatter: one wave per edge (incl. implicit self loops e >= E).
// Lane covers 8 features as two float4 loads + 8 f32 global atomics.
// ---------------------------------------------------------------------------
__global__ __launch_bounds__(256) void gcn_scatter(
    const float* __restrict__ xw, const int* __restrict__ src,
    const int* __restrict__ dst, int E_, int N_,
    const float* __restrict__ isd, float* __restrict__ agg) {
  const int gid  = blockIdx.x * blockDim.x + threadIdx.x;
  const int edge = gid >> 5;
  const int lane = gid & 31;
  const int E2 = E_ + N_;
  if (edge >= E2) return;

  int s, d;
  if (edge < E_) { s = src[edge]; d = dst[edge]; }
  else           { s = d = edge - E_; }           // self loop

  const float norm = isd[s] * isd[d];
  const float4* xr = (const float4*)(xw + (size_t)s * GCN_H);
  float* ar = agg + (size_t)d * GCN_H;

  #pragma unroll
  for (int j = 0; j < 2; ++j) {
    const int c4 = lane + (j << 5);     // float4 index 0..63
    float4 v = xr[c4];
    const int c = c4 << 2;
    unsafeAtomicAdd(&ar[c + 0], v.x * norm);
    unsafeAtomicAdd(&ar[c + 1], v.y * norm);
    unsafeAtomicAdd(&ar[c + 2], v.z * norm);
    unsafeAtomicAdd(&ar[c + 3], v.w * norm);
  }
}

// ---------------------------------------------------------------------------
// Fused bias + BatchNorm(eval) + ReLU:  out = relu((agg+b-m)*g*rsqrt(v+eps)+be)
// ---------------------------------------------------------------------------
__global__ void gcn_bn_relu(
    const float* __restrict__ agg, const float* __restrict__ b,
    const float* __restrict__ g, const float* __restrict__ be,
    const float* __restrict__ m, const float* __restrict__ v,
    float* __restrict__ out, int n) {
  int i = blockIdx.x * blockDim.x + threadIdx.x;
  if (i >= n) return;
  const int f = i & (GCN_H - 1);
  const float scale = g[f] * rsqrtf(v[f] + BN_EPS);
  const float val = (agg[i] + b[f] - m[f]) * scale + be[f];
  out[i] = fmaxf(val, 0.0f);
}

// ---------------------------------------------------------------------------
// Final FC [256 -> 1] + sigmoid. One wave per node; wave32 shfl_xor reduce.
// ---------------------------------------------------------------------------
__global__ __launch_bounds__(256) void gcn_fc_sigmoid(
    const float* __restrict__ h, const float* __restrict__ Wfc,
    const float* __restrict__ bfc, float* __restrict__ out, int n) {
  const int gid  = blockIdx.x * blockDim.x + threadIdx.x;
  const int node = gid >> 5;
  const int lane = gid & 31;
  if (node >= n) return;

  const float4* hr = (const float4*)(h + (size_t)node * GCN_H);
  const float4* wr = (const float4*)Wfc;
  float acc = 0.f;
  #pragma unroll
  for (int j = 0; j < 2; ++j) {
    const int c4 = lane + (j << 5);
    float4 a = hr[c4];
    float4 w = wr[c4];
    acc += a.x * w.x + a.y * w.y + a.z * w.z + a.w * w.w;
  }
  #pragma unroll
  for (int off = 16; off > 0; off >>= 1)
    acc += __shfl_xor(acc, off, 32);
  if (lane == 0) {
    const float z = acc + bfc[0];
    out[node] = 1.0f / (1.0f + expf(-z));
  }
}

// ---------------------------------------------------------------------------
// Host launcher
// ---------------------------------------------------------------------------
extern "C" void kernel_launch(void* const* d_in, const int* in_sizes, int n_in,
                              void* d_out, int out_size, void* d_ws, size_t ws_size,
                              hipStream_t stream) {
  const float* x   = (const float*)d_in[0];
  const int*   ei  = (const int*)d_in[1];
  const float* W1  = (const float*)d_in[2];
  const float* b1  = (const float*)d_in[3];
  const float* W2  = (const float*)d_in[4];
  const float* b2  = (const float*)d_in[5];
  const float* W3  = (const float*)d_in[6];
  const float* b3  = (const float*)d_in[7];
  const float* g1  = (const float*)d_in[8];
  const float* be1 = (const float*)d_in[9];
  const float* m1  = (const float*)d_in[10];
  const float* v1  = (const float*)d_in[11];
  const float* g2  = (const float*)d_in[12];
  const float* be2 = (const float*)d_in[13];
  const float* m2  = (const float*)d_in[14];
  const float* v2  = (const float*)d_in[15];
  const float* g3  = (const float*)d_in[16];
  const float* be3 = (const float*)d_in[17];
  const float* m3  = (const float*)d_in[18];
  const float* v3  = (const float*)d_in[19];
  const float* Wfc = (const float*)d_in[20];
  const float* bfc = (const float*)d_in[21];
  float* out = (float*)d_out;

  const int N = in_sizes[0] / GCN_H;      // 50000
  const int E = in_sizes[1] / 2;          // 800000
  const int* src = ei;
  const int* dst = ei + E;
  const size_t NH = (size_t)N * GCN_H;

  // Workspace layout (all float, 256B-aligned slices)
  float* isd  = (float*)d_ws;                       // N
  float* bufB = isd + ((N + 63) & ~63);             // N*H (xw)
  float* bufC = bufB + NH;                          // N*H (agg)
  float* bufH = bufC + NH;                          // N*H (h)

  const int T = 256;
  const int nBlkN   = (N + T - 1) / T;
  const int nBlkE   = (E + T - 1) / T;
  const int nBlkNH  = (int)((NH + T - 1) / T);
  const int nBlkNH4 = (int)((NH / 4 + T - 1) / T);
  const int E2 = E + N;
  const int nBlkScat = (E2 * 32 + T - 1) / T;       // one wave per edge
  const int nBlkFC   = (N * 32 + T - 1) / T;        // one wave per node
  const int nBlkGemm = N / 16;                      // 16 rows per block (N % 16 == 0)

  // --- degree / normalization ---
  gcn_deg_init<<<nBlkN, T, 0, stream>>>(isd, N);
  gcn_deg_accum<<<nBlkE, T, 0, stream>>>(dst, isd, E);
  gcn_deg_finalize<<<nBlkN, T, 0, stream>>>(isd, N);

  // --- layer 1 ---
  gcn_gemm_wmma<<<nBlkGemm, 512, 0, stream>>>(x, W1, bufB);
  gcn_fill_zero4<<<nBlkNH4, T, 0, stream>>>((float4*)bufC, (int)(NH / 4));
  gcn_scatter<<<nBlkScat, T, 0, stream>>>(bufB, src, dst, E, N, isd, bufC);
  gcn_bn_relu<<<nBlkNH, T, 0, stream>>>(bufC, b1, g1, be1, m1, v1, bufH, (int)NH);

  // --- layer 2 ---
  gcn_gemm_wmma<<<nBlkGemm, 512, 0, stream>>>(bufH, W2, bufB);
  gcn_fill_zero4<<<nBlkNH4, T, 0, stream>>>((float4*)bufC, (int)(NH / 4));
  gcn_scatter<<<nBlkScat, T, 0, stream>>>(bufB, src, dst, E, N, isd, bufC);
  gcn_bn_relu<<<nBlkNH, T, 0, stream>>>(bufC, b2, g2, be2, m2, v2, bufH, (int)NH);

  // --- layer 3 ---
  gcn_gemm_wmma<<<nBlkGemm, 512, 0, stream>>>(bufH, W3, bufB);
  gcn_fill_zero4<<<nBlkNH4, T, 0, stream>>>((float4*)bufC, (int)(NH / 4));
  gcn_scatter<<<nBlkScat, T, 0, stream>>>(bufB, src, dst, E, N, isd, bufC);
  gcn_bn_relu<<<nBlkNH, T, 0, stream>>>(bufC, b3, g3, be3, m3, v3, bufH, (int)NH);

  // --- readout ---
  gcn_fc_sigmoid<<<nBlkFC, T, 0, stream>>>(bufH, Wfc, bfc, out, N);
}